// ShortestPath_67765993996512
// MI455X (gfx1250) — compile-verified
//
#include <hip/hip_runtime.h>
#include <math.h>

// Problem constants (match reference)
#define B_     8
#define N_     256
#define ME_    1024
#define P_     1280   // m + n
#define Q_     512    // 2n
#define ITERS_ 30
#define SIGMA_ 0.1f
#define RIDGE_ 1e-8f
#define BIGV_  1e12f

typedef float v2f __attribute__((ext_vector_type(2)));
typedef float v8f __attribute__((ext_vector_type(8)));

// ---------------------------------------------------------------------------
// Setup: build dense M = [[A, 0],[clip(A,0), I]]  (B x Q x P, row-major)
// ---------------------------------------------------------------------------
__global__ void k_build_M(const float* __restrict__ A, float* __restrict__ M) {
    int t = blockIdx.x * blockDim.x + threadIdx.x;
    if (t >= B_ * Q_ * P_) return;
    int p = t % P_;
    int r2 = t / P_;
    int r = r2 % Q_;
    int b = r2 / Q_;
    float v;
    if (r < N_) {
        v = (p < ME_) ? A[(b * N_ + r) * ME_ + p] : 0.0f;
    } else {
        int rr = r - N_;
        if (p < ME_) {
            float a = A[(b * N_ + rr) * ME_ + p];
            v = a > 0.0f ? a : 0.0f;
        } else {
            v = (p - ME_ == rr) ? 1.0f : 0.0f;
        }
    }
    M[t] = v;
}

__global__ void k_init_vecs(const float* __restrict__ bvec, const float* __restrict__ c,
                            float* __restrict__ z, float* __restrict__ w,
                            float* __restrict__ y, float* __restrict__ rhs,
                            float* __restrict__ cz) {
    int t = blockIdx.x * blockDim.x + threadIdx.x;
    if (t < B_ * P_) {
        int p = t % P_, b = t / P_;
        z[t] = 1.0f;
        w[t] = 1.0f;
        cz[t] = (p < ME_) ? c[b * ME_ + p] : 0.0f;
    }
    if (t < B_ * Q_) {
        int r = t % Q_, b = t / Q_;
        y[t] = 0.0f;
        rhs[t] = (r < N_) ? bvec[b * N_ + r] : 1.0f;
    }
}

// ---------------------------------------------------------------------------
// r_p = M z - rhs        (one wave32 per output row, coalesced row dot)
// ---------------------------------------------------------------------------
__global__ void k_rp(const float* __restrict__ M, const float* __restrict__ z,
                     const float* __restrict__ rhs, float* __restrict__ rp) {
    int wave = (blockIdx.x * blockDim.x + threadIdx.x) >> 5;
    int lane = threadIdx.x & 31;
    if (wave >= B_ * Q_) return;
    int b = wave / Q_, r = wave % Q_;
    const float* Mr = M + (b * Q_ + r) * P_;
    const float* zb = z + b * P_;
    float s = 0.0f;
    for (int k = lane; k < P_; k += 32) s += Mr[k] * zb[k];
    for (int o = 16; o; o >>= 1) s += __shfl_xor(s, o, 32);
    if (lane == 0) rp[wave] = s - rhs[wave];
}

// ---------------------------------------------------------------------------
// r_d = M^T y + w - cz   (thread per column; M[k*P+p] is coalesced across p)
// ---------------------------------------------------------------------------
__global__ void k_rd(const float* __restrict__ M, const float* __restrict__ y,
                     const float* __restrict__ w, const float* __restrict__ cz,
                     float* __restrict__ rd) {
    int t = blockIdx.x * blockDim.x + threadIdx.x;
    if (t >= B_ * P_) return;
    int p = t % P_, b = t / P_;
    const float* Mb = M + b * Q_ * P_;
    const float* yb = y + b * Q_;
    float s = 0.0f;
    for (int k = 0; k < Q_; ++k) s += Mb[k * P_ + p] * yb[k];
    rd[t] = s + w[t] - cz[t];
}

// ---------------------------------------------------------------------------
// mu[b] = mean(z*w)
// ---------------------------------------------------------------------------
__global__ void k_mu(const float* __restrict__ z, const float* __restrict__ w,
                     float* __restrict__ scal) {
    __shared__ float sh[256];
    int b = blockIdx.x, tid = threadIdx.x;
    float s = 0.0f;
    for (int p = tid; p < P_; p += 256) s += z[b * P_ + p] * w[b * P_ + p];
    sh[tid] = s;
    __syncthreads();
    for (int o = 128; o; o >>= 1) {
        if (tid < o) sh[tid] += sh[tid + o];
        __syncthreads();
    }
    if (tid == 0) scal[b * 4 + 0] = sh[0] / (float)P_;
}

// ---------------------------------------------------------------------------
// D = z/w ;  g = D*(r_d + (sigma*mu - z*w)/z)
// ---------------------------------------------------------------------------
__global__ void k_g(const float* __restrict__ z, const float* __restrict__ w,
                    const float* __restrict__ rd, const float* __restrict__ scal,
                    float* __restrict__ D, float* __restrict__ g) {
    int t = blockIdx.x * blockDim.x + threadIdx.x;
    if (t >= B_ * P_) return;
    int b = t / P_;
    float mu = scal[b * 4 + 0];
    float zz = z[t], ww = w[t];
    float comp = (SIGMA_ * mu - zz * ww) / zz;
    float Dv = zz / ww;
    D[t] = Dv;
    g[t] = Dv * (rd[t] + comp);
}

// ---------------------------------------------------------------------------
// rhs_y = -r_p - M g     (wave per row)
// ---------------------------------------------------------------------------
__global__ void k_rhsy(const float* __restrict__ M, const float* __restrict__ g,
                       const float* __restrict__ rp, float* __restrict__ rhsy) {
    int wave = (blockIdx.x * blockDim.x + threadIdx.x) >> 5;
    int lane = threadIdx.x & 31;
    if (wave >= B_ * Q_) return;
    int b = wave / Q_, r = wave % Q_;
    const float* Mr = M + (b * Q_ + r) * P_;
    const float* gb = g + b * P_;
    float s = 0.0f;
    for (int k = lane; k < P_; k += 32) s += Mr[k] * gb[k];
    for (int o = 16; o; o >>= 1) s += __shfl_xor(s, o, 32);
    if (lane == 0) rhsy[wave] = -rp[wave] - s;
}

// ---------------------------------------------------------------------------
// K = (M diag(D)) M^T + ridge*I   via V_WMMA_F32_16X16X4_F32.
// 2x2 register-tiled: one wave computes a 32x32 block of K (4 independent
// accumulators -> independent WMMA chains, 10B loaded per WMMA).
// A-operand layout (16x4 f32): lane 0-15 -> rows, VGPR{0,1} = K{0,1} (lanes<16)
// or K{2,3} (lanes>=16). B-operand (4x16) mirrors it. C/D: VGPR r -> row r / r+8.
// ---------------------------------------------------------------------------
__global__ void k_K(const float* __restrict__ M, const float* __restrict__ D,
                    float* __restrict__ K) {
    int wave = (blockIdx.x * blockDim.x + threadIdx.x) >> 5;
    int lane = threadIdx.x & 31;
    if (wave >= B_ * 16 * 16) return;
    int b = wave / (16 * 16);
    int tt = wave % (16 * 16);
    int ti = tt / 16, tj = tt % 16;       // 32x32 block indices
    int lr = lane & 15, hi = lane >> 4;

    const float* Mb = M + b * Q_ * P_;
    const float* Db = D + b * P_;
    const float* pa0 = Mb + (ti * 32 +      lr) * P_;
    const float* pa1 = Mb + (ti * 32 + 16 + lr) * P_;
    const float* pb0 = Mb + (tj * 32 +      lr) * P_;
    const float* pb1 = Mb + (tj * 32 + 16 + lr) * P_;

    v8f acc00 = {0.f, 0.f, 0.f, 0.f, 0.f, 0.f, 0.f, 0.f};
    v8f acc01 = acc00, acc10 = acc00, acc11 = acc00;

#pragma unroll 4
    for (int k0 = 0; k0 < P_; k0 += 4) {
        int ko = k0 + 2 * hi;
        v2f dv = *(const v2f*)(Db + ko);
        v2f a0 = *(const v2f*)(pa0 + ko);
        v2f a1 = *(const v2f*)(pa1 + ko);
        v2f b0 = *(const v2f*)(pb0 + ko);
        v2f b1 = *(const v2f*)(pb1 + ko);
        a0.x *= dv.x; a0.y *= dv.y;
        a1.x *= dv.x; a1.y *= dv.y;
        acc00 = __builtin_amdgcn_wmma_f32_16x16x4_f32(false, a0, false, b0,
                                                      (short)0, acc00, false, false);
        acc01 = __builtin_amdgcn_wmma_f32_16x16x4_f32(false, a0, false, b1,
                                                      (short)0, acc01, false, false);
        acc10 = __builtin_amdgcn_wmma_f32_16x16x4_f32(false, a1, false, b0,
                                                      (short)0, acc10, false, false);
        acc11 = __builtin_amdgcn_wmma_f32_16x16x4_f32(false, a1, false, b1,
                                                      (short)0, acc11, false, false);
    }

    float* Kb = K + b * Q_ * Q_;
#pragma unroll
    for (int rt = 0; rt < 2; ++rt) {
#pragma unroll
        for (int ct = 0; ct < 2; ++ct) {
            const v8f& acc = rt ? (ct ? acc11 : acc10) : (ct ? acc01 : acc00);
            int col = tj * 32 + ct * 16 + lr;
#pragma unroll
            for (int r = 0; r < 8; ++r) {
                int row = ti * 32 + rt * 16 + r + 8 * hi;
                float v = acc[r];
                if (row == col) v += RIDGE_;
                Kb[row * Q_ + col] = v;
            }
        }
    }
}

// ---------------------------------------------------------------------------
// Per-batch Cholesky (in-place lower) + L L^T dy = rhs_y (RHS kept in LDS)
// ---------------------------------------------------------------------------
__global__ void k_chol(float* __restrict__ K, const float* __restrict__ rhsy,
                       float* __restrict__ dy) {
    __shared__ float svec[Q_];
    __shared__ float sval;
    int b = blockIdx.x, tid = threadIdx.x;
    float* Kb = K + b * Q_ * Q_;

    for (int i = tid; i < Q_; i += 256) svec[i] = rhsy[b * Q_ + i];
    __syncthreads();

    // factor
    for (int k = 0; k < Q_; ++k) {
        if (tid == 0) {
            float d = sqrtf(Kb[k * Q_ + k]);
            Kb[k * Q_ + k] = d;
            sval = 1.0f / d;
        }
        __syncthreads();
        float dinv = sval;
        for (int i = k + 1 + tid; i < Q_; i += 256) Kb[i * Q_ + k] *= dinv;
        __syncthreads();
        for (int i = k + 1 + tid; i < Q_; i += 256) {
            float lik = Kb[i * Q_ + k];
            float* rowi = Kb + i * Q_;
            for (int j = k + 1; j <= i; ++j) rowi[j] -= lik * Kb[j * Q_ + k];
        }
        __syncthreads();
    }
    // forward solve L t = rhs
    for (int k = 0; k < Q_; ++k) {
        if (tid == 0) {
            float t = svec[k] / Kb[k * Q_ + k];
            svec[k] = t;
            sval = t;
        }
        __syncthreads();
        float t = sval;
        for (int i = k + 1 + tid; i < Q_; i += 256) svec[i] -= Kb[i * Q_ + k] * t;
        __syncthreads();
    }
    // back solve L^T x = t   (L^T[i][k] = Kb[k*Q+i], i<k)
    for (int k = Q_ - 1; k >= 0; --k) {
        if (tid == 0) {
            float t = svec[k] / Kb[k * Q_ + k];
            svec[k] = t;
            sval = t;
        }
        __syncthreads();
        float t = sval;
        for (int i = tid; i < k; i += 256) svec[i] -= Kb[k * Q_ + i] * t;
        __syncthreads();
    }
    for (int i = tid; i < Q_; i += 256) dy[b * Q_ + i] = svec[i];
}

// ---------------------------------------------------------------------------
// dz = D * (M^T dy) + g    (thread per column, coalesced)
// ---------------------------------------------------------------------------
__global__ void k_dz(const float* __restrict__ M, const float* __restrict__ dyv,
                     const float* __restrict__ D, const float* __restrict__ g,
                     float* __restrict__ dz) {
    int t = blockIdx.x * blockDim.x + threadIdx.x;
    if (t >= B_ * P_) return;
    int p = t % P_, b = t / P_;
    const float* Mb = M + b * Q_ * P_;
    const float* yb = dyv + b * Q_;
    float s = 0.0f;
    for (int k = 0; k < Q_; ++k) s += Mb[k * P_ + p] * yb[k];
    dz[t] = D[t] * s + g[t];
}

// ---------------------------------------------------------------------------
// dw elementwise + fraction-to-boundary step sizes a_p, a_d (per batch)
// ---------------------------------------------------------------------------
__global__ void k_alpha(const float* __restrict__ z, const float* __restrict__ w,
                        const float* __restrict__ dz, float* __restrict__ dw,
                        float* __restrict__ scal) {
    __shared__ float shp[256], shd[256];
    int b = blockIdx.x, tid = threadIdx.x;
    float mu = scal[b * 4 + 0];
    float mp = BIGV_, md = BIGV_;
    for (int p = tid; p < P_; p += 256) {
        int t = b * P_ + p;
        float zz = z[t], ww = w[t], dzz = dz[t];
        float dwv = (SIGMA_ * mu - zz * ww - ww * dzz) / zz;
        dw[t] = dwv;
        if (dzz < 0.0f) { float r = -zz / dzz; mp = fminf(mp, r); }
        if (dwv < 0.0f) { float r = -ww / dwv; md = fminf(md, r); }
    }
    shp[tid] = mp;
    shd[tid] = md;
    __syncthreads();
    for (int o = 128; o; o >>= 1) {
        if (tid < o) {
            shp[tid] = fminf(shp[tid], shp[tid + o]);
            shd[tid] = fminf(shd[tid], shd[tid + o]);
        }
        __syncthreads();
    }
    if (tid == 0) {
        scal[b * 4 + 1] = fminf(1.0f, 0.99f * shp[0]);
        scal[b * 4 + 2] = fminf(1.0f, 0.99f * shd[0]);
    }
}

// ---------------------------------------------------------------------------
// z += a_p dz ; w += a_d dw ; y += a_d dy
// ---------------------------------------------------------------------------
__global__ void k_update(float* __restrict__ z, float* __restrict__ w,
                         float* __restrict__ y, const float* __restrict__ dz,
                         const float* __restrict__ dw, const float* __restrict__ dyv,
                         const float* __restrict__ scal) {
    int t = blockIdx.x * blockDim.x + threadIdx.x;
    if (t >= B_ * P_) return;
    int p = t % P_, b = t / P_;
    float ap = scal[b * 4 + 1], ad = scal[b * 4 + 2];
    z[t] += ap * dz[t];
    w[t] += ad * dw[t];
    if (p < Q_) y[b * Q_ + p] += ad * dyv[b * Q_ + p];
}

// ---------------------------------------------------------------------------
// out = sum_{b,e} z[b,e] * c[b,e]   (e < m)
// ---------------------------------------------------------------------------
__global__ void k_cost(const float* __restrict__ z, const float* __restrict__ c,
                       float* __restrict__ out) {
    __shared__ float sh[256];
    int tid = threadIdx.x;
    float s = 0.0f;
    for (int t = tid; t < B_ * ME_; t += 256) {
        int e = t % ME_, b = t / ME_;
        s += z[b * P_ + e] * c[t];
    }
    sh[tid] = s;
    __syncthreads();
    for (int o = 128; o; o >>= 1) {
        if (tid < o) sh[tid] += sh[tid + o];
        __syncthreads();
    }
    if (tid == 0) out[0] = sh[0];
}

// ---------------------------------------------------------------------------
extern "C" void kernel_launch(void* const* d_in, const int* in_sizes, int n_in,
                              void* d_out, int out_size, void* d_ws, size_t ws_size,
                              hipStream_t stream) {
    const float* A    = (const float*)d_in[0];  // [B, N, M]
    const float* bvec = (const float*)d_in[1];  // [B, N]
    const float* c    = (const float*)d_in[2];  // [B, M]
    float* out = (float*)d_out;

    // workspace carve-out (256B aligned)
    char* ws = (char*)d_ws;
    size_t off = 0;
    auto alloc = [&](size_t bytes) -> float* {
        float* p = (float*)(ws + off);
        off += (bytes + 255) & ~(size_t)255;
        return p;
    };
    float* Mm   = alloc((size_t)B_ * Q_ * P_ * 4);  // 20 MB
    float* Kw   = alloc((size_t)B_ * Q_ * Q_ * 4);  // 8 MB
    float* z    = alloc((size_t)B_ * P_ * 4);
    float* w    = alloc((size_t)B_ * P_ * 4);
    float* y    = alloc((size_t)B_ * Q_ * 4);
    float* rp   = alloc((size_t)B_ * Q_ * 4);
    float* rhsy = alloc((size_t)B_ * Q_ * 4);
    float* rd   = alloc((size_t)B_ * P_ * 4);
    float* D    = alloc((size_t)B_ * P_ * 4);
    float* g    = alloc((size_t)B_ * P_ * 4);
    float* dz   = alloc((size_t)B_ * P_ * 4);
    float* dw   = alloc((size_t)B_ * P_ * 4);
    float* dyv  = alloc((size_t)B_ * Q_ * 4);
    float* rhs  = alloc((size_t)B_ * Q_ * 4);
    float* cz   = alloc((size_t)B_ * P_ * 4);
    float* scal = alloc((size_t)B_ * 4 * 4);
    (void)ws_size;

    const int TB = 256;
    // setup
    k_build_M<<<(B_ * Q_ * P_ + TB - 1) / TB, TB, 0, stream>>>(A, Mm);
    k_init_vecs<<<(B_ * P_ + TB - 1) / TB, TB, 0, stream>>>(bvec, c, z, w, y, rhs, cz);

    const int rowBlocks  = (B_ * Q_) / 8;        // 8 waves/block, wave per row
    const int colBlocks  = (B_ * P_ + TB - 1) / TB;
    const int tileBlocks = (B_ * 16 * 16) / 8;   // 8 waves/block, wave per 32x32 K block

    for (int it = 0; it < ITERS_; ++it) {
        k_rp<<<rowBlocks, TB, 0, stream>>>(Mm, z, rhs, rp);
        k_rd<<<colBlocks, TB, 0, stream>>>(Mm, y, w, cz, rd);
        k_mu<<<B_, TB, 0, stream>>>(z, w, scal);
        k_g<<<colBlocks, TB, 0, stream>>>(z, w, rd, scal, D, g);
        k_rhsy<<<rowBlocks, TB, 0, stream>>>(Mm, g, rp, rhsy);
        k_K<<<tileBlocks, TB, 0, stream>>>(Mm, D, Kw);
        k_chol<<<B_, TB, 0, stream>>>(Kw, rhsy, dyv);
        k_dz<<<colBlocks, TB, 0, stream>>>(Mm, dyv, D, g, dz);
        k_alpha<<<B_, TB, 0, stream>>>(z, w, dz, dw, scal);
        k_update<<<colBlocks, TB, 0, stream>>>(z, w, y, dz, dw, dyv, scal);
    }
    k_cost<<<1, TB, 0, stream>>>(z, c, out);
    (void)in_sizes; (void)n_in; (void)out_size;
}